// VecInt_35072702939443
// MI455X (gfx1250) — compile-verified
//
#include <hip/hip_runtime.h>

// VecInt: scaling-and-squaring integration of a stationary velocity field.
// vel: [2,128,128,128,3] f32 -> displacement [2,128,128,128,3] f32.
//
// Strategy (MI455X): process batches sequentially so the 48 MB ping-pong
// working set stays resident in the 192 MB L2; HBM sees only the initial
// read and the final write. Each step is a gather-bound pass: 1 streaming
// b96 read, 8 data-dependent b96 gathers (issued together for MLP), 1 b96
// write per voxel. wave32, one voxel per lane.

#define DBITS 7
#define DIM   128
#define NVOX  (DIM * DIM * DIM)        // 2,097,152 voxels per batch

struct F3 { float z, y, x; };          // channel order matches grid (d,h,w)

__device__ __forceinline__ int flat3(int z, int y, int x) {
    return ((((z << DBITS) | y) << DBITS) | x);
}

// dst[p] = scale * ( src[p] + trilerp(src, p + scale*src[p]) )
// scale = 1/2^7 on the first step (folds the initial division into step 1),
// scale = 1 afterwards.
__global__ __launch_bounds__(256)
void vecint_step_kernel(const F3* __restrict__ src, F3* __restrict__ dst,
                        float scale) {
    const int idx = blockIdx.x * blockDim.x + threadIdx.x;
    const int x = idx & (DIM - 1);
    const int y = (idx >> DBITS) & (DIM - 1);
    const int z = idx >> (2 * DBITS);

    const F3 own = src[idx];                       // coalesced b96 stream
    const float vz = own.z * scale;
    const float vy = own.y * scale;
    const float vx = own.x * scale;

    const float mx = (float)(DIM - 1);
    const float lz = fminf(fmaxf((float)z + vz, 0.0f), mx);
    const float ly = fminf(fmaxf((float)y + vy, 0.0f), mx);
    const float lx = fminf(fmaxf((float)x + vx, 0.0f), mx);

    const float fz = floorf(lz), fy = floorf(ly), fx = floorf(lx);
    const int iz0 = (int)fz, iy0 = (int)fy, ix0 = (int)fx;
    const int iz1 = min(iz0 + 1, DIM - 1);
    const int iy1 = min(iy0 + 1, DIM - 1);
    const int ix1 = min(ix0 + 1, DIM - 1);

    const float wz1 = lz - fz, wy1 = ly - fy, wx1 = lx - fx;
    const float wz0 = 1.0f - wz1, wy0 = 1.0f - wy1, wx0 = 1.0f - wx1;

    // Issue all 8 gather loads up front: independent b96 loads -> one wait.
    const F3 g000 = src[flat3(iz0, iy0, ix0)];
    const F3 g001 = src[flat3(iz0, iy0, ix1)];
    const F3 g010 = src[flat3(iz0, iy1, ix0)];
    const F3 g011 = src[flat3(iz0, iy1, ix1)];
    const F3 g100 = src[flat3(iz1, iy0, ix0)];
    const F3 g101 = src[flat3(iz1, iy0, ix1)];
    const F3 g110 = src[flat3(iz1, iy1, ix0)];
    const F3 g111 = src[flat3(iz1, iy1, ix1)];

    const float w000 = wz0 * wy0 * wx0, w001 = wz0 * wy0 * wx1;
    const float w010 = wz0 * wy1 * wx0, w011 = wz0 * wy1 * wx1;
    const float w100 = wz1 * wy0 * wx0, w101 = wz1 * wy0 * wx1;
    const float w110 = wz1 * wy1 * wx0, w111 = wz1 * wy1 * wx1;

    float oz = own.z, oy = own.y, ox = own.x;
    oz = fmaf(w000, g000.z, oz); oy = fmaf(w000, g000.y, oy); ox = fmaf(w000, g000.x, ox);
    oz = fmaf(w001, g001.z, oz); oy = fmaf(w001, g001.y, oy); ox = fmaf(w001, g001.x, ox);
    oz = fmaf(w010, g010.z, oz); oy = fmaf(w010, g010.y, oy); ox = fmaf(w010, g010.x, ox);
    oz = fmaf(w011, g011.z, oz); oy = fmaf(w011, g011.y, oy); ox = fmaf(w011, g011.x, ox);
    oz = fmaf(w100, g100.z, oz); oy = fmaf(w100, g100.y, oy); ox = fmaf(w100, g100.x, ox);
    oz = fmaf(w101, g101.z, oz); oy = fmaf(w101, g101.y, oy); ox = fmaf(w101, g101.x, ox);
    oz = fmaf(w110, g110.z, oz); oy = fmaf(w110, g110.y, oy); ox = fmaf(w110, g110.x, ox);
    oz = fmaf(w111, g111.z, oz); oy = fmaf(w111, g111.y, oy); ox = fmaf(w111, g111.x, ox);

    F3 r;
    r.z = oz * scale;
    r.y = oy * scale;
    r.x = ox * scale;
    dst[idx] = r;                                  // coalesced b96 store
}

extern "C" void kernel_launch(void* const* d_in, const int* in_sizes, int n_in,
                              void* d_out, int out_size, void* d_ws, size_t ws_size,
                              hipStream_t stream) {
    (void)in_sizes; (void)n_in; (void)out_size; (void)ws_size;

    const F3* vel = (const F3*)d_in[0];   // [2, NVOX] of F3
    F3* out = (F3*)d_out;                 // [2, NVOX] of F3
    F3* scr = (F3*)d_ws;                  // 24 MB scratch (one field)

    const float s0 = 1.0f / 128.0f;       // 2^-INT_STEPS, INT_STEPS = 7
    const dim3 blk(256);
    const dim3 grd(NVOX / 256);           // exact cover: 8192 blocks

    for (int b = 0; b < 2; ++b) {
        const F3* v0 = vel + (size_t)b * NVOX;
        F3* O = out + (size_t)b * NVOX;
        F3* S = scr;
        // 7 steps; destinations alternate so step 7 lands in d_out.
        vecint_step_kernel<<<grd, blk, 0, stream>>>(v0, O, s0);    // v1
        vecint_step_kernel<<<grd, blk, 0, stream>>>(O,  S, 1.0f);  // v2
        vecint_step_kernel<<<grd, blk, 0, stream>>>(S,  O, 1.0f);  // v3
        vecint_step_kernel<<<grd, blk, 0, stream>>>(O,  S, 1.0f);  // v4
        vecint_step_kernel<<<grd, blk, 0, stream>>>(S,  O, 1.0f);  // v5
        vecint_step_kernel<<<grd, blk, 0, stream>>>(O,  S, 1.0f);  // v6
        vecint_step_kernel<<<grd, blk, 0, stream>>>(S,  O, 1.0f);  // v7 -> out
    }
}